// BERTFCNCRF_75453985456703
// MI455X (gfx1250) — compile-verified
//
#include <hip/hip_runtime.h>
#include <hip/hip_bf16.h>
#include <math.h>

typedef __attribute__((ext_vector_type(2))) float v2f;
typedef __attribute__((ext_vector_type(8))) float v8f;
typedef __attribute__((ext_vector_type(4))) int   v4i;

#define HDIM 768
#define BATCH 64
#define TLEN 512
#define BT (BATCH * TLEN)          // 32768 tokens
#define WT_STRIDE 770              // 768 + 2 -> LDS bank spread for B-frag reads
#define OUT_STRIDE 17              // 16 + 1 -> conflict-free staging reads
#define XSTRIDE 68                 // 64 + 4 -> conflict-free A-frag reads (banks 4m+2h+k)
#define XBUF (16 * XSTRIDE)        // one 16x64 chunk buffer (1088 floats)
#define NCHUNK (HDIM / 64)         // 12 K-chunks

// ---------------------------------------------------------------------------
// CDNA5 async global->LDS staging (guarded; falls back to direct b64 loads).
// ---------------------------------------------------------------------------
#if defined(__AMDGCN__) && __has_builtin(__builtin_amdgcn_global_load_async_to_lds_b128)
#define USE_ASYNC_STAGE 1
#else
#define USE_ASYNC_STAGE 0
#endif

#if USE_ASYNC_STAGE
typedef __attribute__((address_space(1))) v4i as1_v4i;   // global int4
typedef __attribute__((address_space(3))) v4i as3_v4i;   // LDS int4

__device__ __forceinline__ void async_b128(const float* g, const float* l) {
    // (global src: v4i AS1*, lds dst: v4i AS3*, imm offset, imm cpol)
    __builtin_amdgcn_global_load_async_to_lds_b128(
        (as1_v4i*)(g), (as3_v4i*)(l), 0, 0);
}

template <int N>
__device__ __forceinline__ void wait_asynccnt() {
#if __has_builtin(__builtin_amdgcn_s_wait_asynccnt)
    __builtin_amdgcn_s_wait_asynccnt(N);
#else
    if (N == 0) asm volatile("s_wait_asynccnt 0x0" ::: "memory");
    else        asm volatile("s_wait_asynccnt 0x8" ::: "memory");
#endif
    asm volatile("" ::: "memory");   // keep LDS reads below the wait
}
#endif

// ---------------------------------------------------------------------------
// Kernel 1: fused  z_t = xW_n + b_n ; z_s = xW_u + b_u ; softmax/alpha mix.
// One wave per 16-token tile, V_WMMA_F32_16X16X4_F32 accumulation (fp32 exact).
// Combined weight W[768][16]: cols 0-2 = w_normal, cols 3-9 = w_upper, 10-15 = 0.
// A operand staged via double-buffered async DMA chunks (16 rows x 64 K).
// ---------------------------------------------------------------------------
__global__ __launch_bounds__(256) void k_gemm_mix(
    const float* __restrict__ x,
    const float* __restrict__ w_normal, const float* __restrict__ b_normal,
    const float* __restrict__ w_upper,  const float* __restrict__ b_upper,
    const float* __restrict__ w_trans,
    float* __restrict__ out_tilde,   // d_out + 1 : [BT][7]
    float* __restrict__ zt_buf)      // workspace : [BT][3]
{
    __shared__ float sWT[16 * WT_STRIDE];      // W transposed: sWT[n][k]
    __shared__ float sBias[16];
    __shared__ float sWtr[21];                 // w_trans [3][7]
    __shared__ float sOut[8 * 16 * OUT_STRIDE];
#if USE_ASYNC_STAGE
    __shared__ float sX[8 * 2 * XBUF];         // per-wave double buffer
#endif

    const int tid  = threadIdx.x;
    const int wave = tid >> 5;
    const int lane = tid & 31;
    const int m    = lane & 15;      // A-row / D-col index
    const int half = lane >> 4;      // K-half selector
    const int row0 = (blockIdx.x * 8 + wave) * 16;   // 2048 tiles total, exact

#if USE_ASYNC_STAGE
    // Coalesced DMA mapping: instruction i moves rows {2i+half}, 16B per lane,
    // two contiguous 256B row-segments per issue -> 4 fully-used cachelines.
    float*       myX    = &sX[wave * (2 * XBUF)];
    const float* gstage = x + (size_t)(row0 + half) * HDIM + m * 4;
    float*       lstage = myX + half * XSTRIDE + m * 4;

    // Prefetch chunk 0 into buffer 0 before the sWT cooperative build.
#pragma unroll
    for (int i = 0; i < 8; ++i)
        async_b128(gstage + 2 * i * HDIM, lstage + 2 * i * XSTRIDE);
#endif

    // Build transposed, zero-padded combined weight in LDS.
    for (int idx = tid; idx < 16 * HDIM; idx += 256) {
        int n = idx / HDIM;
        int k = idx - n * HDIM;
        float v = 0.0f;
        if (n < 3)       v = w_normal[k * 3 + n];
        else if (n < 10) v = w_upper [k * 7 + (n - 3)];
        sWT[n * WT_STRIDE + k] = v;
    }
    if (tid < 16) {
        float bv = 0.0f;
        if (tid < 3)       bv = b_normal[tid];
        else if (tid < 10) bv = b_upper [tid - 3];
        sBias[tid] = bv;
    }
    if (tid < 21) sWtr[tid] = w_trans[tid];
    __syncthreads();

    v8f cacc = {};
    // B frag: lane holds W[k + 2*half][m], W[k + 2*half + 1][m]  (via W^T rows)
    const float* bbase0 = &sWT[m * WT_STRIDE + 2 * half];

#if USE_ASYNC_STAGE
    // A frag from LDS: lane holds X[row0+m][k + 2*half .. +1]
    const float* abase0 = myX + m * XSTRIDE + 2 * half;

    for (int ch = 0; ch < NCHUNK; ++ch) {
        if (ch + 1 < NCHUNK) {
            const float* gs = gstage + (ch + 1) * 64;
            float*       ls = lstage + ((ch + 1) & 1) * XBUF;
#pragma unroll
            for (int i = 0; i < 8; ++i)
                async_b128(gs + 2 * i * HDIM, ls + 2 * i * XSTRIDE);
            wait_asynccnt<8>();   // in-order completion: chunk ch is resident
        } else {
            wait_asynccnt<0>();
        }
        const float* ab = abase0 + (ch & 1) * XBUF;
        const float* bb = bbase0 + ch * 64;
#pragma unroll
        for (int kk = 0; kk < 64; kk += 4) {
            v2f a = *(const v2f*)(ab + kk);      // ds_load_b64, conflict-free
            v2f b = *(const v2f*)(bb + kk);      // ds_load_b64, conflict-free
            cacc = __builtin_amdgcn_wmma_f32_16x16x4_f32(
                       false, a, false, b, (short)0, cacc, false, false);
        }
    }
#else
    // Fallback: direct strided global A-frag loads.
    const float* xrow = x + (size_t)(row0 + m) * HDIM + 2 * half;
#pragma unroll 4
    for (int k0 = 0; k0 < HDIM; k0 += 4) {
        v2f a = *(const v2f*)(xrow + k0);
        v2f b = *(const v2f*)(bbase0 + k0);
        cacc = __builtin_amdgcn_wmma_f32_16x16x4_f32(
                   false, a, false, b, (short)0, cacc, false, false);
    }
#endif

    // D layout: lane(m,half), VGPR r -> row M = 8*half + r, col N = m.
    float* so = &sOut[wave * 16 * OUT_STRIDE];
#pragma unroll
    for (int r = 0; r < 8; ++r)
        so[(8 * half + r) * OUT_STRIDE + m] = cacc[r] + sBias[m];
    __syncthreads();

    // Epilogue: one lane per token (lanes 0-15).
    if (lane < 16) {
        const int token = row0 + lane;
        const float* z = &sOut[wave * 16 * OUT_STRIDE + lane * OUT_STRIDE];
        float zt0 = z[0], zt1 = z[1], zt2 = z[2];

        // z_t to workspace for CRF-normal
        zt_buf[(size_t)token * 3 + 0] = zt0;
        zt_buf[(size_t)token * 3 + 1] = zt1;
        zt_buf[(size_t)token * 3 + 2] = zt2;

        // alpha = 0.5 * sum(softmax(z_t)^2)
        float mx = fmaxf(zt0, fmaxf(zt1, zt2));
        float e0 = expf(zt0 - mx), e1 = expf(zt1 - mx), e2 = expf(zt2 - mx);
        float s  = e0 + e1 + e2;
        float alpha = 0.5f * (e0 * e0 + e1 * e1 + e2 * e2) / (s * s);
        float oma = 1.0f - alpha;

        float* ot = out_tilde + (size_t)token * 7;
#pragma unroll
        for (int j = 0; j < 7; ++j) {
            float zsp = zt0 * sWtr[j] + zt1 * sWtr[7 + j] + zt2 * sWtr[14 + j];
            ot[j] = zsp * alpha + z[3 + j] * oma;
        }
    }
}

// ---------------------------------------------------------------------------
// Kernel 2: linear-chain CRF log-likelihood, one single-wave workgroup per
// batch element; lane j owns state j, alpha exchanged via shuffles.
// Writes (score - logZ) per batch to a partial buffer.
// ---------------------------------------------------------------------------
template <int K>
__global__ __launch_bounds__(32) void k_crf(
    const float* __restrict__ E,      // emissions [BATCH][TLEN][K]
    const int*   __restrict__ tags,   // [BATCH][TLEN]
    const int*   __restrict__ mask,   // [BATCH][TLEN]
    const float* __restrict__ trans,  // [K][K]
    const float* __restrict__ start,  // [K]
    const float* __restrict__ endv,   // [K]
    float* __restrict__ partial)
{
    __shared__ float sT[K * K];
    __shared__ float sS[K];
    __shared__ float sE[K];

    const int b    = blockIdx.x;
    const int lane = threadIdx.x;
    for (int i = lane; i < K * K; i += 32) sT[i] = trans[i];
    if (lane < K) { sS[lane] = start[lane]; sE[lane] = endv[lane]; }
    __syncthreads();

    const float* Eb = E    + (size_t)b * TLEN * K;
    const int*   tb = tags + (size_t)b * TLEN;
    const int*   mb = mask + (size_t)b * TLEN;

    const bool act = lane < K;
    const int  jl  = act ? lane : 0;   // clamp for safe LDS indexing

    float e0    = act ? Eb[lane] : 0.0f;
    float alpha = act ? (sS[lane] + e0) : -1e30f;

    const int tag0 = tb[0];
    int   prevTag  = tag0;
    float m0f      = (float)mb[0];
    float emitAcc  = (act && tag0 == lane) ? e0 * m0f : 0.0f;
    float transAcc = 0.0f;
    int   maskSum  = mb[0];

    for (int t = 1; t < TLEN; ++t) {
        const int   mi = mb[t];
        const float mf = (float)mi;
        const int   tg = tb[t];
        const float e  = act ? Eb[(size_t)t * K + lane] : 0.0f;

        float v[K];
        float mx = -3.0e38f;
#pragma unroll
        for (int i = 0; i < K; ++i) {
            float ai = __shfl(alpha, i, 32);
            v[i] = ai + sT[i * K + jl];
            mx = fmaxf(mx, v[i]);
        }
        float se = 0.0f;
#pragma unroll
        for (int i = 0; i < K; ++i) se += expf(v[i] - mx);
        float nxt = mx + logf(se) + e;
        alpha = mi ? nxt : alpha;

        if (act && tg == lane) emitAcc += e * mf;
        if (lane == 0)         transAcc += sT[prevTag * K + tg] * mf;
        prevTag = tg;
        maskSum += mi;
    }

    // logZ = logsumexp_j(alpha_j + end_j)   (all lanes participate in shuffles)
    float term = act ? (alpha + sE[jl]) : -3.0e38f;
    float mx = term;
#pragma unroll
    for (int i = 0; i < K; ++i) mx = fmaxf(mx, __shfl(term, i, 32));
    float sz = 0.0f;
#pragma unroll
    for (int i = 0; i < K; ++i) sz += expf(__shfl(term, i, 32) - mx);
    float logZ = mx + logf(sz);

    float em = 0.0f;
#pragma unroll
    for (int i = 0; i < K; ++i) em += __shfl(emitAcc, i, 32);

    if (lane == 0) {
        int   lastTag = tb[maskSum - 1];
        float score   = em + transAcc + sS[tag0] + sE[lastTag];
        partial[b] = score - logZ;
    }
}

// ---------------------------------------------------------------------------
// Kernel 3: deterministic fixed-order reduction of 128 partials -> loss.
// ---------------------------------------------------------------------------
__global__ void k_reduce(const float* __restrict__ partial, float* __restrict__ out)
{
    if (threadIdx.x == 0 && blockIdx.x == 0) {
        float s = 0.0f;
        for (int i = 0; i < 2 * BATCH; ++i) s += partial[i];
        out[0] = -s;
    }
}

// ---------------------------------------------------------------------------
extern "C" void kernel_launch(void* const* d_in, const int* in_sizes, int n_in,
                              void* d_out, int out_size, void* d_ws, size_t ws_size,
                              hipStream_t stream)
{
    const float* x         = (const float*)d_in[0];
    const int*   attn_mask = (const int*)  d_in[1];
    const int*   labels    = (const int*)  d_in[2];
    const int*   labels_n  = (const int*)  d_in[3];
    const float* w_normal  = (const float*)d_in[4];
    const float* b_normal  = (const float*)d_in[5];
    const float* w_upper   = (const float*)d_in[6];
    const float* b_upper   = (const float*)d_in[7];
    const float* trans_n   = (const float*)d_in[8];
    const float* start_n   = (const float*)d_in[9];
    const float* end_n     = (const float*)d_in[10];
    const float* trans_s   = (const float*)d_in[11];
    const float* start_s   = (const float*)d_in[12];
    const float* end_s     = (const float*)d_in[13];
    const float* w_trans   = (const float*)d_in[14];

    float* out     = (float*)d_out;            // out[0] = loss, out[1..] = z_s_tilde
    float* zt_buf  = (float*)d_ws;             // [BT][3]
    float* partial = zt_buf + (size_t)BT * 3;  // [128]

    // Phase 1: fused GEMM + softmax mix. 2048 token tiles / 8 waves per block.
    k_gemm_mix<<<BT / (16 * 8), 256, 0, stream>>>(
        x, w_normal, b_normal, w_upper, b_upper, w_trans, out + 1, zt_buf);

    // Phase 2: both CRFs (independent; parallel over batch).
    k_crf<3><<<BATCH, 32, 0, stream>>>(zt_buf,  labels_n, attn_mask,
                                       trans_n, start_n, end_n, partial);
    k_crf<7><<<BATCH, 32, 0, stream>>>(out + 1, labels,   attn_mask,
                                       trans_s, start_s, end_s, partial + BATCH);

    // Phase 3: deterministic loss reduction.
    k_reduce<<<1, 32, 0, stream>>>(partial, out);
}